// GCN_10136122819080
// MI455X (gfx1250) — compile-verified
//
#include <hip/hip_runtime.h>
#include <hip/hip_bf16.h>

typedef __attribute__((ext_vector_type(2))) float v2f;
typedef __attribute__((ext_vector_type(8))) float v8f;

#define GCN_N 50000

// ---------------------------------------------------------------- utilities
__global__ void zero_kernel(float* __restrict__ p, long long n) {
    long long i = blockIdx.x * (long long)blockDim.x + threadIdx.x;
    if (i < n) p[i] = 0.0f;
}

__global__ void deg_init_kernel(float* __restrict__ deg, int n) {
    int i = blockIdx.x * blockDim.x + threadIdx.x;
    if (i < n) deg[i] = 1.0f;  // self-loop contributes 1 to in-degree
}

__global__ void deg_accum_kernel(const int* __restrict__ dst, float* __restrict__ deg, int e) {
    int i = blockIdx.x * blockDim.x + threadIdx.x;
    if (i < e) unsafeAtomicAdd(&deg[dst[i]], 1.0f);
}

__global__ void dinv_kernel(float* __restrict__ deg, int n) {
    int i = blockIdx.x * blockDim.x + threadIdx.x;
    if (i < n) deg[i] = rsqrtf(deg[i]);  // deg >= 1 always (self-loops)
}

// Pad W3 [K x nc] into Wp [K x ld] with zero columns nc..ld-1.
__global__ void pad_w_kernel(const float* __restrict__ Wsrc, float* __restrict__ Wp,
                             int K, int nc, int ld) {
    int i = blockIdx.x * blockDim.x + threadIdx.x;
    if (i >= K * ld) return;
    int k = i / ld, c = i % ld;
    Wp[i] = (c < nc) ? Wsrc[k * nc + c] : 0.0f;
}

// ---------------------------------------------------------------- GEMM (fp32 WMMA)
// C[M x NCOLS] = A[M x K] * W[K x NCOLS], row-major. K, NCOLS compile-time,
// both multiples of 16 (W pre-padded for layer 3); M is a multiple of 16.
// Each wave computes TWO 16-row tiles across NT 16-wide column tiles, so the
// NT B-fragment loads per k-step feed 2*NT WMMAs. No guards in the K loop:
// every load is unconditional -> clean pipelining, single waits, WMMA bursts.
// Fragment layouts per CDNA5 ISA 7.12.2:
//   A 16x4 f32 : lanes 0-15 hold M=lane, K={0,1}; lanes 16-31 hold M=lane-16, K={2,3}
//   B 4x16 f32 : lanes 0-15 hold N=lane, K={0,1}; lanes 16-31 hold N=lane-16, K={2,3}
//   C/D 16x16  : VGPR v -> row v (lanes 0-15) / row v+8 (lanes 16-31), col = lane&15
template <int NT, int K, int NCOLS>
__global__ __launch_bounds__(256) void gemm_f32_wmma(
    const float* __restrict__ A, const float* __restrict__ W,
    float* __restrict__ C, int M) {
    const int lane     = threadIdx.x & 31;
    const int wave     = threadIdx.x >> 5;
    const int pairIdx  = blockIdx.x * 8 + wave;
    const int rowBase0 = pairIdx * 32;
    if (rowBase0 >= M) return;                    // wave-uniform
    const int  rowBase1 = rowBase0 + 16;
    const bool has1     = (rowBase1 < M);         // wave-uniform (M % 16 == 0)

    const int mlane = lane & 15;
    const int khalf = lane >> 4;                  // 0 or 1
    const int r0 = rowBase0 + mlane;
    const int r1 = has1 ? (rowBase1 + mlane) : r0;  // clamp: no OOB, result unused

    const float* arow0 = A + (size_t)r0 * K + 2 * khalf;
    const float* arow1 = A + (size_t)r1 * K + 2 * khalf;
    const float* wbase = W + (size_t)(2 * khalf) * NCOLS + mlane;

    v8f acc0[NT] = {};
    v8f acc1[NT] = {};

#pragma unroll 2
    for (int k0 = 0; k0 < K; k0 += 4) {
        const v2f a0 = *(const v2f*)(arow0 + k0);
        const v2f a1 = *(const v2f*)(arow1 + k0);
        v2f b[NT];
#pragma unroll
        for (int t = 0; t < NT; ++t) {
            const float* wp = wbase + (size_t)k0 * NCOLS + t * 16;
            b[t].x = wp[0];
            b[t].y = wp[NCOLS];
        }
#pragma unroll
        for (int t = 0; t < NT; ++t) {
            acc0[t] = __builtin_amdgcn_wmma_f32_16x16x4_f32(
                false, a0, false, b[t], (short)0, acc0[t], false, false);
            acc1[t] = __builtin_amdgcn_wmma_f32_16x16x4_f32(
                false, a1, false, b[t], (short)0, acc1[t], false, false);
        }
    }

#pragma unroll
    for (int t = 0; t < NT; ++t)
#pragma unroll
        for (int v = 0; v < 8; ++v)
            C[(size_t)(rowBase0 + v + 8 * khalf) * NCOLS + t * 16 + mlane] = acc0[t][v];

    if (has1) {                                   // wave-uniform branch
#pragma unroll
        for (int t = 0; t < NT; ++t)
#pragma unroll
            for (int v = 0; v < 8; ++v)
                C[(size_t)(rowBase1 + v + 8 * khalf) * NCOLS + t * 16 + mlane] = acc1[t][v];
    }
}

// ---------------------------------------------------------------- edge scatter-add
// One wave per edge; lanes stride over features.
__global__ void scatter_kernel(const float* __restrict__ h, const int* __restrict__ src,
                               const int* __restrict__ dst, const float* __restrict__ dinv,
                               float* __restrict__ agg, int e, int F, int ld) {
    const int warp = (int)((blockIdx.x * (size_t)blockDim.x + threadIdx.x) >> 5);
    const int lane = threadIdx.x & 31;
    if (warp >= e) return;
    const int s = src[warp];
    const int d = dst[warp];
    const float w = dinv[s] * dinv[d];
    const float* hs = h + (size_t)s * ld;
    float* ad = agg + (size_t)d * ld;
    for (int f = lane; f < F; f += 32)
        unsafeAtomicAdd(&ad[f], hs[f] * w);
}

// ---------------------------------------------------------------- finalize: self-loop + bias + relu (F=128)
__global__ void finalize_relu_kernel(float* __restrict__ agg, const float* __restrict__ h,
                                     const float* __restrict__ dinv, const float* __restrict__ bias,
                                     int n) {
    const long long idx = blockIdx.x * (long long)blockDim.x + threadIdx.x;
    if (idx >= (long long)n << 7) return;
    const int i = (int)(idx >> 7);
    const int f = (int)(idx & 127);
    const float di = dinv[i];
    const float v = agg[idx] + h[idx] * di * di + bias[f];
    agg[idx] = v > 0.0f ? v : 0.0f;
}

// ---------------------------------------------------------------- finalize: self-loop + bias + log_softmax
// One wave per node row; C <= 64 classes handled by two lanesworth of columns.
__global__ void finalize_lsm_kernel(const float* __restrict__ agg, const float* __restrict__ h,
                                    const float* __restrict__ dinv, const float* __restrict__ bias,
                                    float* __restrict__ out, int n, int C, int ld) {
    const int row  = (int)((blockIdx.x * (size_t)blockDim.x + threadIdx.x) >> 5);
    const int lane = threadIdx.x & 31;
    if (row >= n) return;
    const float di = dinv[row];
    const float d2 = di * di;
    const size_t base = (size_t)row * ld;

    float x0 = -1e30f, x1 = -1e30f;
    if (lane < C)      x0 = agg[base + lane]      + h[base + lane]      * d2 + bias[lane];
    if (lane + 32 < C) x1 = agg[base + lane + 32] + h[base + lane + 32] * d2 + bias[lane + 32];

    float m = fmaxf(x0, x1);
    for (int off = 16; off; off >>= 1) m = fmaxf(m, __shfl_xor(m, off, 32));

    float s = 0.0f;
    if (lane < C)      s += expf(x0 - m);
    if (lane + 32 < C) s += expf(x1 - m);
    for (int off = 16; off; off >>= 1) s += __shfl_xor(s, off, 32);

    const float lse = m + logf(s);
    if (lane < C)      out[(size_t)row * C + lane]      = x0 - lse;
    if (lane + 32 < C) out[(size_t)row * C + lane + 32] = x1 - lse;
}

// ---------------------------------------------------------------- driver
extern "C" void kernel_launch(void* const* d_in, const int* in_sizes, int n_in,
                              void* d_out, int out_size, void* d_ws, size_t ws_size,
                              hipStream_t stream) {
    const float* x  = (const float*)d_in[0];
    const int*   ei = (const int*)d_in[1];   // [2, E] flat: src then dst
    const float* W1 = (const float*)d_in[2];
    const float* b1 = (const float*)d_in[3];
    const float* W2 = (const float*)d_in[4];
    const float* b2 = (const float*)d_in[5];
    const float* W3 = (const float*)d_in[6];
    const float* b3 = (const float*)d_in[7];
    float* out = (float*)d_out;

    const int N = GCN_N;
    const int E = in_sizes[1] / 2;
    const int NC = 40, LD3 = 48;

    // workspace layout (floats)
    float* dinv  = (float*)d_ws;                 // N
    float* bufA  = dinv + N;                     // N*128   (GEMM output h)
    float* bufB  = bufA + (size_t)N * 128;       // N*128   (aggregation / next input)
    float* bufC  = bufB + (size_t)N * 128;       // N*48    (layer-3 h, padded)
    float* bufD  = bufC + (size_t)N * LD3;       // N*48    (layer-3 agg, padded)
    float* W3pad = bufD + (size_t)N * LD3;       // 128*48  (zero-padded W3)

    const int pairTiles     = (N + 31) / 32;            // 32-row wave strips
    const int gemmBlocks    = (pairTiles + 7) / 8;      // 8 waves/block
    const int scatterBlocks = (E + 7) / 8;              // 8 edges(waves)/block of 256
    const long long nf      = (long long)N * 128;
    const int nfBlocks      = (int)((nf + 255) / 256);

    // --- degrees -> dinv ---
    deg_init_kernel<<<(N + 255) / 256, 256, 0, stream>>>(dinv, N);
    deg_accum_kernel<<<(E + 255) / 256, 256, 0, stream>>>(ei + E, dinv, E);
    dinv_kernel<<<(N + 255) / 256, 256, 0, stream>>>(dinv, N);

    // --- pad W3 once (deterministic, part of every launch) ---
    pad_w_kernel<<<(128 * LD3 + 255) / 256, 256, 0, stream>>>(W3, W3pad, 128, NC, LD3);

    // --- layer 1: h = x @ W1 ; agg ; relu(agg + self + b1) ---
    gemm_f32_wmma<8, 256, 128><<<gemmBlocks, 256, 0, stream>>>(x, W1, bufA, N);
    zero_kernel<<<nfBlocks, 256, 0, stream>>>(bufB, nf);
    scatter_kernel<<<scatterBlocks, 256, 0, stream>>>(bufA, ei, ei + E, dinv, bufB, E, 128, 128);
    finalize_relu_kernel<<<nfBlocks, 256, 0, stream>>>(bufB, bufA, dinv, b1, N);

    // --- layer 2 ---
    gemm_f32_wmma<8, 128, 128><<<gemmBlocks, 256, 0, stream>>>(bufB, W2, bufA, N);
    zero_kernel<<<nfBlocks, 256, 0, stream>>>(bufB, nf);
    scatter_kernel<<<scatterBlocks, 256, 0, stream>>>(bufA, ei, ei + E, dinv, bufB, E, 128, 128);
    finalize_relu_kernel<<<nfBlocks, 256, 0, stream>>>(bufB, bufA, dinv, b2, N);

    // --- layer 3: 40 classes (W padded to 48 cols), then log_softmax ---
    const long long nf3 = (long long)N * LD3;
    gemm_f32_wmma<3, 128, 48><<<gemmBlocks, 256, 0, stream>>>(bufB, W3pad, bufC, N);
    zero_kernel<<<(int)((nf3 + 255) / 256), 256, 0, stream>>>(bufD, nf3);
    scatter_kernel<<<scatterBlocks, 256, 0, stream>>>(bufC, ei, ei + E, dinv, bufD, E, NC, LD3);
    finalize_lsm_kernel<<<(N * 32 + 255) / 256, 256, 0, stream>>>(bufD, bufC, dinv, b3, out, N, NC, LD3);
}